// Decoder_14121852469560
// MI455X (gfx1250) — compile-verified
//
#include <hip/hip_runtime.h>
#include <hip/hip_bf16.h>

// ---------------------------------------------------------------------------
// Pointer-generator decoder step for MI455X (gfx1250, wave32, WMMA).
// Dominant cost: vocab GEMM streaming W_vocab (614 MB) -> HBM bound.
// All GEMMs use V_WMMA_F32_16X16X4_F32 (exact fp32 matrix path on CDNA5).
// ---------------------------------------------------------------------------

typedef __attribute__((ext_vector_type(2))) float v2f;
typedef __attribute__((ext_vector_type(8))) float v8f;

static constexpr int Bq   = 64;
static constexpr int Eq   = 512;
static constexpr int Hq   = 1024;
static constexpr int Sq   = 512;
static constexpr int Tq   = 32;
static constexpr int Vq   = 50000;
static constexpr int OOVq = 50;
static constexpr int H4   = 4 * Hq;   // 4096
static constexpr int H3   = 3 * Hq;   // 3072
static constexpr int VEXT = Vq + OOVq; // 50050

__device__ __forceinline__ v8f wmma4(v2f a, v2f b, v8f c) {
  // D = A(16x4 f32) * B(4x16 f32) + C(16x16 f32)
  return __builtin_amdgcn_wmma_f32_16x16x4_f32(false, a, false, b, (short)0, c,
                                               false, false);
}
__device__ __forceinline__ float sigm(float x) { return 1.0f / (1.0f + expf(-x)); }

// ---------------------------------------------------------------------------
// K1: gates[64,4096] = outputs @ W_ih^T + enc_h @ W_hh^T + b_ih + b_hh
// One wave per 16x16 tile. grid = (4096/16, 64/16)
// ---------------------------------------------------------------------------
__global__ __launch_bounds__(32) void k_gates(
    const float* __restrict__ X, const float* __restrict__ Hh,
    const float* __restrict__ Wih, const float* __restrict__ Whh,
    const float* __restrict__ bih, const float* __restrict__ bhh,
    float* __restrict__ gates) {
  const int lane = threadIdx.x;
  const int l = lane & 15;
  const int hf = lane >> 4;           // which K-half this lane holds
  const int n0 = blockIdx.x * 16;
  const int m0 = blockIdx.y * 16;

  v8f acc0 = {}, acc1 = {};
  {
    const float* arow = X + (size_t)(m0 + l) * Eq + 2 * hf;
    const float* brow = Wih + (size_t)(n0 + l) * Eq + 2 * hf;
    for (int k = 0; k < Eq; k += 8) {
      v2f a0 = *(const v2f*)(arow + k);
      v2f b0 = *(const v2f*)(brow + k);
      v2f a1 = *(const v2f*)(arow + k + 4);
      v2f b1 = *(const v2f*)(brow + k + 4);
      acc0 = wmma4(a0, b0, acc0);
      acc1 = wmma4(a1, b1, acc1);
    }
  }
  {
    const float* arow = Hh + (size_t)(m0 + l) * Hq + 2 * hf;
    const float* brow = Whh + (size_t)(n0 + l) * Hq + 2 * hf;
    for (int k = 0; k < Hq; k += 8) {
      v2f a0 = *(const v2f*)(arow + k);
      v2f b0 = *(const v2f*)(brow + k);
      v2f a1 = *(const v2f*)(arow + k + 4);
      v2f b1 = *(const v2f*)(brow + k + 4);
      acc0 = wmma4(a0, b0, acc0);
      acc1 = wmma4(a1, b1, acc1);
    }
  }
  v8f acc = acc0 + acc1;
  const float bias = bih[n0 + l] + bhh[n0 + l];
  float* col = gates + (size_t)(m0 + 8 * hf) * H4 + (n0 + l);
#pragma unroll
  for (int r = 0; r < 8; ++r) col[(size_t)r * H4] = acc[r] + bias;
}

// ---------------------------------------------------------------------------
// K2: LSTM pointwise (gate order i,f,g,o). 65536 threads.
// ---------------------------------------------------------------------------
__global__ void k_lstm(const float* __restrict__ gates,
                       const float* __restrict__ enc_c,
                       float* __restrict__ out_h, float* __restrict__ out_cell,
                       float* __restrict__ out_nph_last) {
  int idx = blockIdx.x * blockDim.x + threadIdx.x;  // b*H + j
  int b = idx >> 10, j = idx & (Hq - 1);
  const float* g = gates + (size_t)b * H4;
  float gi = g[j], gf = g[Hq + j], gg = g[2 * Hq + j], go = g[3 * Hq + j];
  float c = sigm(gf) * enc_c[idx] + sigm(gi) * tanhf(gg);
  float h = sigm(go) * tanhf(c);
  out_cell[idx] = c;
  out_h[idx] = h;
  out_nph_last[idx] = h;  // new_previous_hidden[T] slice
}

// one-float4-per-thread copy (previous_hidden -> new_previous_hidden[0:T])
__global__ void k_copy4(const float4* __restrict__ src, float4* __restrict__ dst,
                        int n4) {
  int i = blockIdx.x * blockDim.x + threadIdx.x;
  if (i < n4) dst[i] = src[i];
}

// ---------------------------------------------------------------------------
// K3: scores[b,s] = exp(sum_h h[b,h] * enc[b,h,s]); coalesced over s.
// ---------------------------------------------------------------------------
__global__ void k_scores(const float* __restrict__ h,
                         const float* __restrict__ enc,
                         float* __restrict__ scores) {
  int idx = blockIdx.x * blockDim.x + threadIdx.x;  // 32768
  int b = idx >> 9, s = idx & (Sq - 1);
  const float* e = enc + (size_t)b * Hq * Sq + s;
  const float* hb = h + (size_t)b * Hq;
  float acc = 0.0f;
#pragma unroll 4
  for (int j = 0; j < Hq; ++j) acc += hb[j] * e[(size_t)j * Sq];
  scores[idx] = expf(acc);
}

// K4: per-b attention normalization; writes new_scores_sum + temporal_attention
__global__ void k_attn_norm(float* __restrict__ scores,  // in: scores, out: attn
                            const float* __restrict__ tss,
                            float* __restrict__ out_nss,
                            float* __restrict__ out_ta) {
  __shared__ float red[256];
  int b = blockIdx.x, tid = threadIdx.x;
  float local = 0.0f;
  for (int s = tid; s < Sq; s += 256) {
    float sc = scores[b * Sq + s];
    float t = tss[b * Sq + s];
    float at = sc / t;
    out_nss[b * Sq + s] = sc + t;
    scores[b * Sq + s] = at;
    local += at;
  }
  red[tid] = local;
  __syncthreads();
  for (int off = 128; off; off >>= 1) {
    if (tid < off) red[tid] += red[tid + off];
    __syncthreads();
  }
  float inv = 1.0f / red[0];
  for (int s = tid; s < Sq; s += 256) out_ta[b * Sq + s] = scores[b * Sq + s] * inv;
}

// K5: ctx_enc[b,h] = sum_s ta[b,s] * enc[b,h,s]; ta staged in LDS.
__global__ void k_ctx_enc(const float* __restrict__ ta,
                          const float* __restrict__ enc,
                          float* __restrict__ ctx) {
  __shared__ float sta[Sq];
  int b = blockIdx.x, tid = threadIdx.x;
  for (int s = tid; s < Sq; s += 256) sta[s] = ta[b * Sq + s];
  __syncthreads();
  for (int hh = tid; hh < Hq; hh += 256) {
    const float* e = enc + (size_t)b * Hq * Sq + (size_t)hh * Sq;
    float acc = 0.0f;
#pragma unroll 4
    for (int s = 0; s < Sq; ++s) acc += sta[s] * e[s];
    ctx[b * Hq + hh] = acc;
  }
}

// ---------------------------------------------------------------------------
// K6: proj = previous_hidden(viewed [T*B=2048, H]) @ W_attn^T. WMMA f32.
// grid = (1024/16, 2048/16)
// ---------------------------------------------------------------------------
__global__ __launch_bounds__(32) void k_proj(const float* __restrict__ prev,
                                             const float* __restrict__ Wa,
                                             float* __restrict__ proj) {
  const int lane = threadIdx.x;
  const int l = lane & 15;
  const int hf = lane >> 4;
  const int n0 = blockIdx.x * 16;
  const int m0 = blockIdx.y * 16;
  const float* arow = prev + (size_t)(m0 + l) * Hq + 2 * hf;
  const float* brow = Wa + (size_t)(n0 + l) * Hq + 2 * hf;
  v8f acc0 = {}, acc1 = {};
  for (int k = 0; k < Hq; k += 8) {
    v2f a0 = *(const v2f*)(arow + k);
    v2f b0 = *(const v2f*)(brow + k);
    v2f a1 = *(const v2f*)(arow + k + 4);
    v2f b1 = *(const v2f*)(brow + k + 4);
    acc0 = wmma4(a0, b0, acc0);
    acc1 = wmma4(a1, b1, acc1);
  }
  v8f acc = acc0 + acc1;
  float* col = proj + (size_t)(m0 + 8 * hf) * Hq + (n0 + l);
#pragma unroll
  for (int r = 0; r < 8; ++r) col[(size_t)r * Hq] = acc[r];
}

// K7: logits[b,t] = h[b,:] . proj[t,b,:]   (one wave per output)
__global__ __launch_bounds__(32) void k_logits(const float* __restrict__ h,
                                               const float* __restrict__ proj,
                                               float* __restrict__ logits) {
  int id = blockIdx.x;  // b*T + t
  int b = id / Tq, t = id % Tq;
  int lane = threadIdx.x;
  const float* hb = h + (size_t)b * Hq;
  const float* pr = proj + (size_t)(t * Bq + b) * Hq;
  float acc = 0.0f;
  for (int j = lane; j < Hq; j += 32) acc += hb[j] * pr[j];
  for (int m = 16; m; m >>= 1) acc += __shfl_xor(acc, m, 32);
  if (lane == 0) logits[id] = acc;
}

// K8: softmax over T=32 (one wave) + ctx_dec[b,h] = sum_t a[t]*prev[t,b,h]
__global__ void k_dec(const float* __restrict__ logits,
                      const float* __restrict__ prev, float* __restrict__ ctxd) {
  __shared__ float sa[Tq];
  int b = blockIdx.x, tid = threadIdx.x;
  if (tid < 32) {
    float v = logits[b * Tq + tid];
    float m = v;
    for (int off = 16; off; off >>= 1) m = fmaxf(m, __shfl_xor(m, off, 32));
    float e = expf(v - m);
    float s2 = e;
    for (int off = 16; off; off >>= 1) s2 += __shfl_xor(s2, off, 32);
    sa[tid] = e / s2;
  }
  __syncthreads();
  for (int hh = tid; hh < Hq; hh += 256) {
    float acc = 0.0f;
#pragma unroll
    for (int t = 0; t < Tq; ++t) acc += sa[t] * prev[(size_t)(t * Bq + b) * Hq + hh];
    ctxd[b * Hq + hh] = acc;
  }
}

// K9: feats = [h | ctx_enc | ctx_dec]
__global__ void k_feats(const float* __restrict__ h, const float* __restrict__ ce,
                        const float* __restrict__ cd, float* __restrict__ feats) {
  int idx = blockIdx.x * blockDim.x + threadIdx.x;  // 65536
  int b = idx >> 10, j = idx & (Hq - 1);
  feats[(size_t)b * H3 + j] = h[idx];
  feats[(size_t)b * H3 + Hq + j] = ce[idx];
  feats[(size_t)b * H3 + 2 * Hq + j] = cd[idx];
}

// K10: p[b] = sigmoid(feats[b,:] . W_ptr + b_ptr)
__global__ void k_ptr(const float* __restrict__ feats, const float* __restrict__ Wp,
                      const float* __restrict__ bp, float* __restrict__ p) {
  __shared__ float red[256];
  int b = blockIdx.x, tid = threadIdx.x;
  float local = 0.0f;
  for (int j = tid; j < H3; j += 256) local += feats[(size_t)b * H3 + j] * Wp[j];
  red[tid] = local;
  __syncthreads();
  for (int off = 128; off; off >>= 1) {
    if (tid < off) red[tid] += red[tid + off];
    __syncthreads();
  }
  if (tid == 0) p[b] = sigm(red[0] + bp[0]);
}

// ---------------------------------------------------------------------------
// K11: vocab logits [64,50000] = feats @ W_vocab^T + b_vocab. Dominant kernel.
// One wave computes a 64-row x 16-col strip (4 M-tiles reuse each B load), so
// W_vocab (614 MB) streams from HBM exactly once. 50000/16 = 3125 waves.
// ---------------------------------------------------------------------------
__global__ __launch_bounds__(32) void k_vocab(const float* __restrict__ feats,
                                              const float* __restrict__ Wv,
                                              const float* __restrict__ bv,
                                              float* __restrict__ vlog) {
  const int lane = threadIdx.x;
  const int l = lane & 15;
  const int hf = lane >> 4;
  const int n0 = blockIdx.x * 16;
  const float* a0 = feats + (size_t)(0 + l) * H3 + 2 * hf;
  const float* a1 = feats + (size_t)(16 + l) * H3 + 2 * hf;
  const float* a2 = feats + (size_t)(32 + l) * H3 + 2 * hf;
  const float* a3 = feats + (size_t)(48 + l) * H3 + 2 * hf;
  const float* brow = Wv + (size_t)(n0 + l) * H3 + 2 * hf;
  v8f acc0 = {}, acc1 = {}, acc2 = {}, acc3 = {};
  for (int k = 0; k < H3; k += 4) {
    v2f b = *(const v2f*)(brow + k);
    v2f x0 = *(const v2f*)(a0 + k);
    v2f x1 = *(const v2f*)(a1 + k);
    v2f x2 = *(const v2f*)(a2 + k);
    v2f x3 = *(const v2f*)(a3 + k);
    acc0 = wmma4(x0, b, acc0);
    acc1 = wmma4(x1, b, acc1);
    acc2 = wmma4(x2, b, acc2);
    acc3 = wmma4(x3, b, acc3);
  }
  const float bias = bv[n0 + l];
  float* col = vlog + (size_t)(8 * hf) * Vq + (n0 + l);
#pragma unroll
  for (int r = 0; r < 8; ++r) col[(size_t)r * Vq] = acc0[r] + bias;
  col += (size_t)16 * Vq;
#pragma unroll
  for (int r = 0; r < 8; ++r) col[(size_t)r * Vq] = acc1[r] + bias;
  col += (size_t)16 * Vq;
#pragma unroll
  for (int r = 0; r < 8; ++r) col[(size_t)r * Vq] = acc2[r] + bias;
  col += (size_t)16 * Vq;
#pragma unroll
  for (int r = 0; r < 8; ++r) col[(size_t)r * Vq] = acc3[r] + bias;
}

// K12: per-b softmax over V, scale by (1-p), write final[b, 0:V]; zero OOV tail
__global__ void k_vsm(const float* __restrict__ vlog, const float* __restrict__ p,
                      float* __restrict__ fin) {
  __shared__ float red[256];
  int b = blockIdx.x, tid = threadIdx.x;
  const float* row = vlog + (size_t)b * Vq;
  float m = -1e30f;
  for (int v = tid; v < Vq; v += 256) m = fmaxf(m, row[v]);
  red[tid] = m;
  __syncthreads();
  for (int off = 128; off; off >>= 1) {
    if (tid < off) red[tid] = fmaxf(red[tid], red[tid + off]);
    __syncthreads();
  }
  float mx = red[0];
  __syncthreads();
  float s = 0.0f;
  for (int v = tid; v < Vq; v += 256) s += expf(row[v] - mx);
  red[tid] = s;
  __syncthreads();
  for (int off = 128; off; off >>= 1) {
    if (tid < off) red[tid] += red[tid + off];
    __syncthreads();
  }
  float scale = (1.0f - p[b]) / red[0];
  float* out = fin + (size_t)b * VEXT;
  for (int v = tid; v < VEXT; v += 256)
    out[v] = (v < Vq) ? expf(row[v] - mx) * scale : 0.0f;
}

// K13: scatter-add copy distribution: final[b, tex[s,b]] += p[b]*ta[b,s]
__global__ void k_scatter(const float* __restrict__ ta, const float* __restrict__ p,
                          const int* __restrict__ tex, float* __restrict__ fin) {
  int idx = blockIdx.x * blockDim.x + threadIdx.x;  // 32768
  int b = idx & (Bq - 1), s = idx / Bq;
  float val = p[b] * ta[b * Sq + s];
  int col = tex[s * Bq + b];
  atomicAdd(fin + (size_t)b * VEXT + col, val);
}

// ---------------------------------------------------------------------------
extern "C" void kernel_launch(void* const* d_in, const int* in_sizes, int n_in,
                              void* d_out, int out_size, void* d_ws,
                              size_t ws_size, hipStream_t stream) {
  const float* outputs = (const float*)d_in[0];
  const float* enc_out = (const float*)d_in[1];
  const float* enc_h   = (const float*)d_in[2];
  const float* enc_c   = (const float*)d_in[3];
  const float* tss     = (const float*)d_in[4];
  const float* prev    = (const float*)d_in[5];
  const float* Wih     = (const float*)d_in[6];
  const float* Whh     = (const float*)d_in[7];
  const float* bih     = (const float*)d_in[8];
  const float* bhh     = (const float*)d_in[9];
  const float* Wattn   = (const float*)d_in[10];
  const float* Wvocab  = (const float*)d_in[11];
  const float* bvocab  = (const float*)d_in[12];
  const float* Wptr    = (const float*)d_in[13];
  const float* bptr    = (const float*)d_in[14];
  const int*   tex     = (const int*)d_in[15];

  // Output tuple packed flat in return order.
  float* out    = (float*)d_out;
  float* o_fin  = out;                         // [64, 50050]
  float* o_h    = out + (size_t)Bq * VEXT;     // [64, 1024]
  float* o_cell = o_h + Bq * Hq;               // [64, 1024]
  float* o_ta   = o_cell + Bq * Hq;            // [64, 1, 512]
  float* o_nss  = o_ta + Bq * Sq;              // [64, 1, 512]
  float* o_nph  = o_nss + Bq * Sq;             // [33, 64, 1024]

  // Workspace layout (floats): ~23.7 MB total.
  float* ws       = (float*)d_ws;
  float* w_gates  = ws;                        // 64*4096   = 262144
  float* w_scores = w_gates + Bq * H4;         // 64*512    = 32768
  float* w_ctxe   = w_scores + Bq * Sq;        // 64*1024   = 65536
  float* w_proj   = w_ctxe + Bq * Hq;          // 2048*1024 = 2097152
  float* w_logt   = w_proj + (size_t)Tq * Bq * Hq;  // 2048
  float* w_ctxd   = w_logt + Bq * Tq;          // 65536
  float* w_feats  = w_ctxd + Bq * Hq;          // 64*3072   = 196608
  float* w_p      = w_feats + Bq * H3;         // 64
  float* w_vlog   = w_p + Bq;                  // 64*50000  = 3200000

  const dim3 wave(32);

  k_gates<<<dim3(H4 / 16, Bq / 16), wave, 0, stream>>>(outputs, enc_h, Wih, Whh,
                                                       bih, bhh, w_gates);
  k_lstm<<<256, 256, 0, stream>>>(w_gates, enc_c, o_h, o_cell,
                                  o_nph + (size_t)Tq * Bq * Hq);
  k_copy4<<<2048, 256, 0, stream>>>((const float4*)prev, (float4*)o_nph,
                                    Tq * Bq * Hq / 4);
  k_scores<<<128, 256, 0, stream>>>(o_h, enc_out, w_scores);
  k_attn_norm<<<Bq, 256, 0, stream>>>(w_scores, tss, o_nss, o_ta);
  k_ctx_enc<<<Bq, 256, 0, stream>>>(o_ta, enc_out, w_ctxe);
  k_proj<<<dim3(Hq / 16, Tq * Bq / 16), wave, 0, stream>>>(prev, Wattn, w_proj);
  k_logits<<<Bq * Tq, wave, 0, stream>>>(o_h, w_proj, w_logt);
  k_dec<<<Bq, 256, 0, stream>>>(w_logt, prev, w_ctxd);
  k_feats<<<256, 256, 0, stream>>>(o_h, w_ctxe, w_ctxd, w_feats);
  k_ptr<<<Bq, 256, 0, stream>>>(w_feats, Wptr, bptr, w_p);
  k_vocab<<<Vq / 16, wave, 0, stream>>>(w_feats, Wvocab, bvocab, w_vlog);
  k_vsm<<<Bq, 256, 0, stream>>>(w_vlog, w_p, o_fin);
  k_scatter<<<128, 256, 0, stream>>>(o_ta, w_p, tex, o_fin);
}